// SELayer3D_85383949844625
// MI455X (gfx1250) — compile-verified
//
#include <hip/hip_runtime.h>
#include <math.h>

#define C 128
#define Bq 8

typedef __attribute__((ext_vector_type(2))) float v2f;
typedef __attribute__((ext_vector_type(4))) float v4f;
typedef __attribute__((ext_vector_type(8))) float v8f;

// workspace layout (floats): [0,1024) segment sums, [1024,1032) counts, [1032,2056) gate
#define WS_SUMS 0
#define WS_CNT  1024
#define WS_GATE 1032
#define WS_TOT  2056

__global__ void se_zero_ws(float* __restrict__ ws) {
    for (int i = threadIdx.x; i < WS_TOT; i += blockDim.x) ws[i] = 0.0f;
}

// ---------------------------------------------------------------------------
// Pass 1: segment sums + counts.  256 threads = 8 rows x 32 float4 lanes.
// bidx is sorted, so each thread keeps a running accumulator for its current
// batch and flushes to LDS atomics only on batch change (<= 7 flushes).
// ---------------------------------------------------------------------------
__global__ __launch_bounds__(256) void se_segsum(const float* __restrict__ x,
                                                 const long long* __restrict__ coors,
                                                 float* __restrict__ ws, int N) {
    __shared__ float ssum[Bq * C];
    __shared__ float scnt[Bq];
    const int tid = threadIdx.x;
    for (int i = tid; i < Bq * C; i += 256) ssum[i] = 0.0f;
    if (tid < Bq) scnt[tid] = 0.0f;
    __syncthreads();

    const int cg   = tid & 31;   // float4 column group (4*cg .. 4*cg+3)
    const int rsub = tid >> 5;   // row 0..7 within block tile
    const long long stride = (long long)gridDim.x * 8;

    int cur = -1;
    float ax = 0.f, ay = 0.f, az = 0.f, aw = 0.f, rc = 0.f;

    for (long long row = (long long)blockIdx.x * 8 + rsub; row < N; row += stride) {
        const int b = (int)coors[row * 4];   // batch id (low word of int64)
        if (b != cur) {
            if (cur >= 0) {
                float* dst = &ssum[cur * C + cg * 4];
                atomicAdd(dst + 0, ax); atomicAdd(dst + 1, ay);
                atomicAdd(dst + 2, az); atomicAdd(dst + 3, aw);
                if (cg == 0) atomicAdd(&scnt[cur], rc);
            }
            cur = b; ax = ay = az = aw = 0.f; rc = 0.f;
        }
        const v4f v = ((const v4f*)(x + row * (long long)C))[cg];
        ax += v.x; ay += v.y; az += v.z; aw += v.w; rc += 1.0f;
    }
    if (cur >= 0) {
        float* dst = &ssum[cur * C + cg * 4];
        atomicAdd(dst + 0, ax); atomicAdd(dst + 1, ay);
        atomicAdd(dst + 2, az); atomicAdd(dst + 3, aw);
        if (cg == 0) atomicAdd(&scnt[cur], rc);
    }
    __syncthreads();
    for (int i = tid; i < Bq * C; i += 256) atomicAdd(&ws[WS_SUMS + i], ssum[i]);
    if (tid < Bq) atomicAdd(&ws[WS_CNT + tid], scnt[tid]);
}

// ---------------------------------------------------------------------------
// Pass 2: gate MLP on a single wave using V_WMMA_F32_16X16X4_F32.
//   GEMM1: mean(16x128, rows 8..15 zero) @ W1p(128x16, cols 8..15 zero)
//          -> 32 K-steps accumulating one 16x16 f32 tile.
//   GEMM2: h(16x8) @ W2(8x128) -> 8 N-tiles x 2 K-steps.
// f32 WMMA lane layout (ISA 7.12.2):
//   A 16x4 : lane L (half=L>>4, m=L&15): vgpr0=A[m][k0+2*half], vgpr1=A[m][k0+2*half+1]
//   B 4x16 : lane L (n=L&15):            vgpr0=B[k0+2*half][n], vgpr1=B[k0+2*half+1][n]
//   C 16x16: vgpr r, lane L:             C[r + 8*half][n]
// ---------------------------------------------------------------------------
__global__ __launch_bounds__(32) void se_gate(const float* __restrict__ W1,
                                              const float* __restrict__ W2,
                                              float* __restrict__ ws) {
    __shared__ float sMean[16 * C];   // padded mean, rows 8..15 = 0
    __shared__ float sW1[C * 16];     // padded W1,  cols 8..15 = 0
    __shared__ float sW2[Bq * C];
    __shared__ float sH[16 * 16];     // relu(mean@W1) tile

    const int L = threadIdx.x;
    const float* sums = ws + WS_SUMS;
    const float* cnts = ws + WS_CNT;
    float* gate = ws + WS_GATE;

    for (int i = L; i < 16 * C; i += 32) {
        const int m = i >> 7, k = i & 127;
        float v = 0.0f;
        if (m < Bq) v = sums[m * C + k] / fmaxf(cnts[m], 1.0f);
        sMean[i] = v;
    }
    for (int i = L; i < C * 16; i += 32) {
        const int k = i >> 4, n = i & 15;
        sW1[i] = (n < Bq) ? W1[k * Bq + n] : 0.0f;
    }
    for (int i = L; i < Bq * C; i += 32) sW2[i] = W2[i];
    __syncthreads();

    const int half = L >> 4;
    const int n = L & 15;   // B/C column for this lane
    const int m = L & 15;   // A row for this lane

    // GEMM1: c1 += mean[:, k0:k0+4] x W1p[k0:k0+4, :]
    v8f c1 = {};
    for (int k0 = 0; k0 < C; k0 += 4) {
        const int ka = k0 + 2 * half;
        v2f a, b;
        a[0] = sMean[m * C + ka];
        a[1] = sMean[m * C + ka + 1];
        b[0] = sW1[ka * 16 + n];
        b[1] = sW1[(ka + 1) * 16 + n];
        c1 = __builtin_amdgcn_wmma_f32_16x16x4_f32(false, a, false, b,
                                                   (short)0, c1, false, false);
    }
    // ReLU -> sH (16x16; rows 8..15 are zero because mean rows 8..15 are zero)
#pragma unroll
    for (int r = 0; r < 8; ++r) {
        sH[(r + 8 * half) * 16 + n] = fmaxf(c1[r], 0.0f);
    }
    __syncthreads();

    // GEMM2: one 16-wide N tile at a time, K = 8 (2 steps)
    for (int t = 0; t < 8; ++t) {
        v8f c2 = {};
        for (int k0 = 0; k0 < 8; k0 += 4) {
            const int ka = k0 + 2 * half;
            v2f a, b;
            a[0] = sH[m * 16 + ka];
            a[1] = sH[m * 16 + ka + 1];
            b[0] = sW2[ka * C + t * 16 + n];
            b[1] = sW2[(ka + 1) * C + t * 16 + n];
            c2 = __builtin_amdgcn_wmma_f32_16x16x4_f32(false, a, false, b,
                                                       (short)0, c2, false, false);
        }
        if (half == 0) {
#pragma unroll
            for (int r = 0; r < 8; ++r) {
                const float z = c2[r];
                gate[r * C + t * 16 + n] = 1.0f / (1.0f + __expf(-z));
            }
        }
    }
}

// ---------------------------------------------------------------------------
// Pass 3: out[n, :] = x[n, :] * gate[bidx[n], :].  Gate (4 KB) lives in LDS.
// v4f (b128) loads; non-temporal b128 stores (out is never re-read).
// ---------------------------------------------------------------------------
__global__ __launch_bounds__(256) void se_scale(const float* __restrict__ x,
                                                const long long* __restrict__ coors,
                                                const float* __restrict__ gate,
                                                float* __restrict__ out, int N) {
    __shared__ float sg[Bq * C];
    const int tid = threadIdx.x;
    for (int i = tid; i < Bq * C; i += 256) sg[i] = gate[i];
    __syncthreads();

    const int cg   = tid & 31;
    const int rsub = tid >> 5;
    const long long stride = (long long)gridDim.x * 8;

    for (long long row = (long long)blockIdx.x * 8 + rsub; row < N; row += stride) {
        const int b = (int)coors[row * 4];
        const v4f v = ((const v4f*)(x + row * (long long)C))[cg];
        const v4f g = ((const v4f*)(sg + b * C))[cg];
        v4f o;
        o.x = v.x * g.x; o.y = v.y * g.y; o.z = v.z * g.z; o.w = v.w * g.w;
        __builtin_nontemporal_store(o, ((v4f*)(out + row * (long long)C)) + cg);
    }
}

extern "C" void kernel_launch(void* const* d_in, const int* in_sizes, int n_in,
                              void* d_out, int out_size, void* d_ws, size_t ws_size,
                              hipStream_t stream) {
    const float*     x     = (const float*)d_in[0];
    const float*     W1    = (const float*)d_in[1];
    const float*     W2    = (const float*)d_in[2];
    const long long* coors = (const long long*)d_in[3];
    float* ws  = (float*)d_ws;
    float* out = (float*)d_out;
    const int N = in_sizes[0] / C;

    se_zero_ws<<<1, 256, 0, stream>>>(ws);
    se_segsum<<<1024, 256, 0, stream>>>(x, coors, ws, N);
    se_gate<<<1, 32, 0, stream>>>(W1, W2, ws);
    se_scale<<<2048, 256, 0, stream>>>(x, coors, ws + WS_GATE, out, N);
}